// InterleavedDepthwiseConv_38663295599003
// MI455X (gfx1250) — compile-verified
//
#include <hip/hip_runtime.h>

typedef float v2f __attribute__((ext_vector_type(2)));
typedef float v8f __attribute__((ext_vector_type(8)));

#define BATCH 2
#define CIN   24
#define DDIM  48
#define HDIM  128
#define WDIM  128
#define NG    3
#define GS    12       // channels per group (in and out)
#define GOUT  36       // NG * GS
#define TAPS  27       // 3x3x3
#define HT    2        // output rows per block
#define HROWS (HT + 2) // staged input rows (with halo)
#define WTILE 130      // WDIM + 2 halo
#define IN_TILE_FLOATS (GS * 3 * HROWS * WTILE)  // 12*3*4*130 = 18720
#define WMAT_ROWS      (TAPS * GS)               // 324 (multiple of 4)
#define WMAT_FLOATS    (WMAT_ROWS * 16)          // 5184 (N padded 12 -> 16)

__global__ __launch_bounds__(256)
void InterleavedDepthwiseConv_wmma(const float* __restrict__ x,
                                   const float* __restrict__ weight,
                                   const float* __restrict__ bias,
                                   float* __restrict__ out) {
  extern __shared__ float lds[];
  float* in_tile = lds;                   // [c][dz][dyrow 0..3][wx 0..129]
  float* w_tile  = lds + IN_TILE_FLOATS;  // [k = t*12 + c][n], n in [0,16)

  const int blk = blockIdx.x;
  const int h0 = (blk % (HDIM / HT)) * HT;         // first of HT output rows
  const int d  = (blk / (HDIM / HT)) % DDIM;
  const int g  = (blk / ((HDIM / HT) * DDIM)) % NG;
  const int b  = blk / ((HDIM / HT) * DDIM * NG);
  const int cstart = g * 6;                        // overlapping slices: 0, 6, 12

  const int tid = threadIdx.x;

  // ---- stage weights: w_tile[(t*12+c)*16 + n] = W[g][n][c][t] ----
  for (int i = tid; i < WMAT_FLOATS; i += 256) {
    const int n = i & 15;
    const int k = i >> 4;
    const int t = k / GS;
    const int c = k - t * GS;
    float v = 0.0f;
    if (n < GS) v = weight[(((size_t)g * GS + n) * GS + c) * TAPS + t];
    w_tile[i] = v;
  }

  // ---- stage input slab (12 ch x 3 d x 4 h x 130 w, zero-padded halo) ----
  for (int i = tid; i < IN_TILE_FLOATS; i += 256) {
    const int wx = i % WTILE;
    const int r  = i / WTILE;
    const int dy = r % HROWS;
    const int r2 = r / HROWS;
    const int dz = r2 % 3;
    const int c  = r2 / 3;
    const int dd = d + dz - 1;
    const int hh = h0 + dy - 1;
    const int ww = wx - 1;
    float v = 0.0f;
    if ((unsigned)dd < DDIM && (unsigned)hh < HDIM && (unsigned)ww < WDIM)
      v = x[(((size_t)(b * CIN + cstart + c) * DDIM + dd) * HDIM + hh) * WDIM + ww];
    in_tile[i] = v;
  }

  __syncthreads();

  const int lane = tid & 31;     // wave32
  const int wave = tid >> 5;     // 0..7
  const int row  = wave >> 2;    // output row within block: 0..1
  const int w0   = (wave & 3) * 32;  // this wave owns W tiles [w0,w0+16) and [w0+16,w0+32)
  const int m    = lane & 15;    // A row / D column select
  const int hi   = lane >> 4;    // lane-half: K pair {0,1} vs {2,3}
  const int n    = m;            // B/D column = out channel

  v8f acc0 = {0.f, 0.f, 0.f, 0.f, 0.f, 0.f, 0.f, 0.f};
  v8f acc1 = {0.f, 0.f, 0.f, 0.f, 0.f, 0.f, 0.f, 0.f};

  const int aw0 = w0 + m;        // spatial base, tile 0
  const int aw1 = aw0 + 16;      // spatial base, tile 1

  // K = 324 in chunks of 4 channels per tap: 81 x (1 B-frag, 2 A-frags, 2 WMMAs)
  #pragma unroll
  for (int t = 0; t < TAPS; ++t) {
    const int dz = t / 9, dyt = (t % 9) / 3, dx = t % 3;
    #pragma unroll
    for (int cc = 0; cc < 3; ++cc) {
      const int c0 = cc * 4 + hi * 2;  // this lane-half's channel pair base
      // A fragments (16x4 f32): VGPR0/1 hold the K pair for this lane-half
      const int rb0 = ((c0 * 3 + dz) * HROWS + row + dyt) * WTILE + dx;
      const int rb1 = (((c0 + 1) * 3 + dz) * HROWS + row + dyt) * WTILE + dx;
      v2f a0, a1;
      a0.x = in_tile[rb0 + aw0];
      a0.y = in_tile[rb1 + aw0];
      a1.x = in_tile[rb0 + aw1];
      a1.y = in_tile[rb1 + aw1];
      // Shared B fragment (4x16 f32): rows k = t*12 + c0, k+1; column n
      const int kb = t * GS + c0;
      v2f bf;
      bf.x = w_tile[kb * 16 + n];
      bf.y = w_tile[(kb + 1) * 16 + n];
      acc0 = __builtin_amdgcn_wmma_f32_16x16x4_f32(
          false, a0, false, bf, (short)0, acc0, false, false);
      acc1 = __builtin_amdgcn_wmma_f32_16x16x4_f32(
          false, a1, false, bf, (short)0, acc1, false, false);
    }
  }

  // ---- bias + store: lane owns out-channel n; D rows M = r + 8*hi -> w = wbase + M
  if (n < GS) {
    const float bv = bias[g * GS + n];
    const size_t obase =
        ((size_t)(b * GOUT + g * GS + n) * DDIM + d) * (size_t)(HDIM * WDIM) +
        (size_t)(h0 + row) * WDIM;
    #pragma unroll
    for (int r = 0; r < 8; ++r) {
      const int mrow = r + 8 * hi;
      out[obase + w0 + mrow]      = acc0[r] + bv;
      out[obase + w0 + 16 + mrow] = acc1[r] + bv;
    }
  }
}

extern "C" void kernel_launch(void* const* d_in, const int* in_sizes, int n_in,
                              void* d_out, int out_size, void* d_ws, size_t ws_size,
                              hipStream_t stream) {
  (void)in_sizes; (void)n_in; (void)out_size; (void)d_ws; (void)ws_size;
  const float* x      = (const float*)d_in[0];
  const float* weight = (const float*)d_in[1];
  const float* bias   = (const float*)d_in[2];
  float* out          = (float*)d_out;

  const dim3 grid(BATCH * NG * DDIM * (HDIM / HT));  // 18432 blocks: (b,g,d,h-pair)
  const dim3 block(256);                             // 8 wave32 waves
  const size_t shmem = (IN_TILE_FLOATS + WMAT_FLOATS) * sizeof(float);  // 95.6 KB

  hipLaunchKernelGGL(InterleavedDepthwiseConv_wmma, grid, block, shmem, stream,
                     x, weight, bias, out);
}